// CLabelSmoothingCrossEntropyLoss_10316511445103
// MI455X (gfx1250) — compile-verified
//
#include <hip/hip_runtime.h>
#include <hip/hip_bf16.h>

// CLabelSmoothingCrossEntropyLoss for MI455X (gfx1250).
//
// Roofline: x is 65536x1000 f32 = 262 MB streamed once -> ~11.3 us at 23.3 TB/s.
// Pure bandwidth problem; WMMA recast would be ~1000x excess FLOPs (131 GFLOP
// dense for 65M useful MACs) and is deliberately absent. CDNA5-specific paths
// used instead:
//   - GLOBAL_LOAD_ASYNC_TO_LDS_B128 (ASYNCcnt) double-buffered gather of the
//     L2-resident smoothing row S[target], one full row of compute ahead.
//   - ds_swizzle_b32 xor-butterfly wave32 reductions (1 instr/step).
//   - non-temporal b128 streaming loads for single-use x rows.

#define CLASSES        1000
#define SROW_STRIDE    1024   // S row padded to 4096 B: exact async-B128 tiling, zero tail
#define BATCH_N        65536
#define SMOOTH         0.1f
#define ROWS_PER_BLOCK 32
#define NBLK_B         (BATCH_N / ROWS_PER_BLOCK)   // 2048

typedef float v4f __attribute__((ext_vector_type(4)));

// ds_swizzle_b32 group-of-32 xor mode: imm = xor_mask<<10 | or<<5 | and(0x1f)
#define SWZF(v, imm) __int_as_float(__builtin_amdgcn_ds_swizzle(__float_as_int(v), (imm)))

__device__ __forceinline__ float wave_sum(float v) {
  v += SWZF(v, 0x401f);   // xor 16
  v += SWZF(v, 0x201f);   // xor 8
  v += SWZF(v, 0x101f);   // xor 4
  v += SWZF(v, 0x081f);   // xor 2
  v += SWZF(v, 0x041f);   // xor 1
  return v;
}
__device__ __forceinline__ float wave_max(float v) {
  v = fmaxf(v, SWZF(v, 0x401f));
  v = fmaxf(v, SWZF(v, 0x201f));
  v = fmaxf(v, SWZF(v, 0x101f));
  v = fmaxf(v, SWZF(v, 0x081f));
  v = fmaxf(v, SWZF(v, 0x041f));
  return v;
}

// 256-thread (8 wave32) block reductions. red[] needs 16 floats.
__device__ __forceinline__ float block_max_bcast(float v, float* red) {
  const int lane = threadIdx.x & 31, w = threadIdx.x >> 5;
  v = wave_max(v);
  if (lane == 0) red[w] = v;
  __syncthreads();
  if (threadIdx.x == 0) {
    float r = red[0];
#pragma unroll
    for (int i = 1; i < 8; ++i) r = fmaxf(r, red[i]);
    red[0] = r;
  }
  __syncthreads();
  const float out = red[0];
  __syncthreads();
  return out;
}
__device__ __forceinline__ float block_sum_bcast(float v, float* red) {
  const int lane = threadIdx.x & 31, w = threadIdx.x >> 5;
  v = wave_sum(v);
  if (lane == 0) red[w] = v;
  __syncthreads();
  if (threadIdx.x == 0) {
    float r = red[0];
#pragma unroll
    for (int i = 1; i < 8; ++i) r += red[i];
    red[0] = r;
  }
  __syncthreads();
  const float out = red[0];
  __syncthreads();
  return out;
}
// Two sums in one barrier round; results valid on thread 0 only.
__device__ __forceinline__ float2 block_sum2_t0(float a, float b, float* red) {
  const int lane = threadIdx.x & 31, w = threadIdx.x >> 5;
  a = wave_sum(a);
  b = wave_sum(b);
  if (lane == 0) { red[w] = a; red[8 + w] = b; }
  __syncthreads();
  float2 out = {0.f, 0.f};
  if (threadIdx.x == 0) {
    float ra = red[0], rb = red[8];
#pragma unroll
    for (int i = 1; i < 8; ++i) { ra += red[i]; rb += red[8 + i]; }
    out.x = ra; out.y = rb;
  }
  __syncthreads();   // red safely reusable next round
  return out;
}

// ---------------------------------------------------------------------------
// Kernel A: S[t,j] = SMOOTH*(1-A[t,j])/(1-A[t,t]), S[t,t] = 1-SMOOTH,
// A = softmax(class_avg). Rows stored with stride 1024, pad zeroed.
// ---------------------------------------------------------------------------
__global__ void build_smoothing_kernel(const float* __restrict__ ca,
                                       float* __restrict__ S,
                                       float* __restrict__ rowsumS) {
  __shared__ float red[16];
  const int t   = blockIdx.x;
  const int tid = threadIdx.x;
  const bool valid = tid < 250;                 // 250*4 = 1000
  const float* row = ca + (size_t)t * CLASSES;  // 4000 B rows -> 16B aligned

  v4f c = {0.f, 0.f, 0.f, 0.f};
  if (valid) c = *(const v4f*)(row + tid * 4);

  float m = valid ? fmaxf(fmaxf(c.x, c.y), fmaxf(c.z, c.w)) : -__builtin_inff();
  m = block_max_bcast(m, red);

  v4f e = {0.f, 0.f, 0.f, 0.f};
  float zp = 0.f;
  if (valid) {
    e.x = expf(c.x - m); e.y = expf(c.y - m);
    e.z = expf(c.z - m); e.w = expf(c.w - m);
    zp = e.x + e.y + e.z + e.w;
  }
  const float Z    = block_sum_bcast(zp, red);
  const float invZ = 1.0f / Z;
  const float Att  = expf(row[t] - m) * invZ;   // uniform scalar load
  const float scal = SMOOTH / (1.0f - Att);

  v4f s = {0.f, 0.f, 0.f, 0.f};
  float rs = 0.f;
  if (valid) {
    s.x = scal * (1.0f - e.x * invZ);
    s.y = scal * (1.0f - e.y * invZ);
    s.z = scal * (1.0f - e.z * invZ);
    s.w = scal * (1.0f - e.w * invZ);
    const int d = t - tid * 4;
    if (d >= 0 && d < 4) ((float*)&s)[d] = 1.0f - SMOOTH;
    rs = s.x + s.y + s.z + s.w;
  }
  const float2 r2 = block_sum2_t0(rs, 0.f, red);

  ((v4f*)(S + (size_t)t * SROW_STRIDE))[tid] = s;   // tid>=250 writes the zero pad
  if (tid == 0) rowsumS[t] = r2.x;
}

// ---------------------------------------------------------------------------
// Kernel B: loss_i = lse_i * rowsumS[t_i] - dot(x_i, S[t_i]).
// Double-buffered async S gather, one row ahead of compute.
// ---------------------------------------------------------------------------
#define ISSUE_S_GATHER(ldsaddr, tgtv)                                          \
  do {                                                                         \
    const unsigned long long _g =                                              \
        (unsigned long long)(size_t)(S + (size_t)(tgtv) * SROW_STRIDE + tid * 4); \
    asm volatile("global_load_async_to_lds_b128 %0, %1, off"                   \
                 :: "v"(ldsaddr), "v"(_g) : "memory");                         \
  } while (0)

__global__ void loss_kernel(const float* __restrict__ x,
                            const int* __restrict__ target,
                            const float* __restrict__ S,
                            const float* __restrict__ rowsumS,
                            float* __restrict__ partials) {
  __shared__ v4f   stage[2][256];   // ping-pong 2 x 4 KB for gathered S rows
  __shared__ float red[16];
  const int tid    = threadIdx.x;
  const bool valid = tid < 250;
  const int row0   = blockIdx.x * ROWS_PER_BLOCK;

  // Generic->LDS: low 32 bits of a __shared__ generic address are the LDS offset.
  const unsigned lds_a[2] = { (unsigned)(size_t)(&stage[0][tid]),
                              (unsigned)(size_t)(&stage[1][tid]) };

  float acc = 0.0f;
  int tgt = target[row0];            // uniform -> scalar load
  ISSUE_S_GATHER(lds_a[0], tgt);     // prime the pipeline

#pragma unroll 1
  for (int r = 0; r < ROWS_PER_BLOCK; ++r) {
    const int row = row0 + r;
    const float rsum = rowsumS[tgt];         // uniform scalar load
    const bool more = (r + 1 < ROWS_PER_BLOCK);

    int tgt_next = tgt;
    if (more) {                              // prefetch next S row into other buffer
      tgt_next = target[row + 1];
      ISSUE_S_GATHER(lds_a[(r + 1) & 1], tgt_next);
    }

    // Stream this sample's logits (single-use -> non-temporal, keeps S in L2).
    v4f xv = {0.f, 0.f, 0.f, 0.f};
    if (valid)
      xv = __builtin_nontemporal_load((const v4f*)(x + (size_t)row * CLASSES) + tid);

    // log-sum-exp stats while the DMA is in flight.
    float m = valid ? fmaxf(fmaxf(xv.x, xv.y), fmaxf(xv.z, xv.w)) : -__builtin_inff();
    m = block_max_bcast(m, red);
    float zp = 0.f;
    if (valid)
      zp = expf(xv.x - m) + expf(xv.y - m) + expf(xv.z - m) + expf(xv.w - m);

    // Async loads complete in order: <=1 outstanding means the current row's
    // copy landed. Each wave reads exactly the LDS bytes its own op wrote.
    if (more) asm volatile("s_wait_asynccnt 0x1" ::: "memory");
    else      asm volatile("s_wait_asynccnt 0x0" ::: "memory");

    const v4f sv = stage[r & 1][tid];
    const float dp = xv.x * sv.x + xv.y * sv.y + xv.z * sv.z + xv.w * sv.w;

    const float2 zd = block_sum2_t0(zp, dp, red);   // {Z, dot} in one round
    if (tid == 0) acc += (m + logf(zd.x)) * rsum - zd.y;

    tgt = tgt_next;
  }
  if (tid == 0) partials[blockIdx.x] = acc;
}

// ---------------------------------------------------------------------------
// Kernel C: deterministic final reduction -> mean.
// ---------------------------------------------------------------------------
__global__ void finalize_kernel(const float* __restrict__ partials,
                                float* __restrict__ out) {
  __shared__ float red[16];
  float v = 0.0f;
  for (int i = threadIdx.x; i < NBLK_B; i += 256) v += partials[i];
  const float2 r2 = block_sum2_t0(v, 0.f, red);
  if (threadIdx.x == 0) out[0] = r2.x * (1.0f / (float)BATCH_N);
}

extern "C" void kernel_launch(void* const* d_in, const int* in_sizes, int n_in,
                              void* d_out, int out_size, void* d_ws, size_t ws_size,
                              hipStream_t stream) {
  const float* x      = (const float*)d_in[0];   // [65536, 1000] f32
  const float* ca     = (const float*)d_in[1];   // [1000, 1000] f32
  const int*   target = (const int*)d_in[2];     // [65536]

  float* S        = (float*)d_ws;                       // 1000*1024 floats (4 MB)
  float* rowsumS  = S + (size_t)SROW_STRIDE * CLASSES;  // 1000 (padded slot 1024)
  float* partials = rowsumS + SROW_STRIDE;              // 2048 floats

  build_smoothing_kernel<<<CLASSES, 256, 0, stream>>>(ca, S, rowsumS);
  loss_kernel<<<NBLK_B, 256, 0, stream>>>(x, target, S, rowsumS, partials);
  finalize_kernel<<<1, 256, 0, stream>>>(partials, (float*)d_out);
}